// MultiTaskLossWrapper_87076166959144
// MI455X (gfx1250) — compile-verified
//
#include <hip/hip_runtime.h>
#include <hip/hip_bf16.h>
#include <math.h>

typedef __attribute__((ext_vector_type(2))) float v2f;
typedef __attribute__((ext_vector_type(8))) float v8f;

#define BATCH 256
#define KDIM  128   // C*M = 8*16
#define NDIM  4096
#define LEN_R 64
#define LEN_D 64

// Kernel 1: outputs[b,n] = sum_k W[k] * x[b,k,n], via V_WMMA_F32_16X16X4_F32.
// Grid: (NDIM / (8 waves * 16 cols) = 32, BATCH). Block: 256 threads = 8 waves.
// Each wave computes one 16-column tile of one batch row; K=128 consumed as 32
// chained f32 WMMAs with W replicated across all 16 M-rows of A.
__global__ void __launch_bounds__(256)
conv_einsum_wmma_kernel(const float* __restrict__ x,
                        const float* __restrict__ W,
                        float* __restrict__ out) {
    __shared__ float wlds[KDIM];
    const int tid = threadIdx.x;
    if (tid < KDIM) wlds[tid] = W[tid];
    __syncthreads();

    const int lane = tid & 31;
    const int wave = tid >> 5;
    const int b    = blockIdx.y;
    const int nt   = (blockIdx.x * 8 + wave) * 16;
    const int col  = nt + (lane & 15);
    const int sel  = lane >> 4;           // 0: half-wave holds K0/K1, 1: K2/K3

    const float* xb = x + (size_t)b * KDIM * NDIM + col;

    v8f c = {};
#pragma unroll 4
    for (int kb = 0; kb < KDIM / 4; ++kb) {
        const int k0 = kb * 4 + sel * 2;
        // A (16x4 f32): lanes0-15 -> {K0,K1}, lanes16-31 -> {K2,K3}; all M rows = W
        v2f a;
        a.x = wlds[k0];
        a.y = wlds[k0 + 1];
        // B (4x16 f32): v0 holds rows K0(lo half)/K2(hi half), v1 holds K1/K3
        v2f bm;
        bm.x = xb[(size_t)k0 * NDIM];
        bm.y = xb[(size_t)(k0 + 1) * NDIM];
        // 8 args: (neg_a, A, neg_b, B, c_mod, C, reuse_a, reuse_b)
        c = __builtin_amdgcn_wmma_f32_16x16x4_f32(false, a, false, bm,
                                                  (short)0, c, false, false);
    }

    // D row M=0: VGPR0 of C on lanes 0-15, N = lane.
    if (lane < 16) out[(size_t)b * NDIM + col] = c[0];
}

// Kernel 2: per-b argmax over 4096 (first-occurrence tie-break, like
// jnp.argmax), then r/d lookups with trunc-toward-zero (int cast semantics).
// Output tail layout (after outputs): d[256], r[256], d[256].
__global__ void __launch_bounds__(256)
argmax_rd_kernel(const float* __restrict__ outs,
                 const float* __restrict__ r_t,
                 const float* __restrict__ d_t,
                 float* __restrict__ tail) {
    __shared__ float sv[256];
    __shared__ int   si[256];
    const int b   = blockIdx.x;
    const int tid = threadIdx.x;
    const float* row = outs + (size_t)b * NDIM;

    float best = -INFINITY;
    int   bi   = 0;
    for (int i = tid; i < NDIM; i += 256) {
        const float v = row[i];
        if (v > best) { best = v; bi = i; }
    }
    sv[tid] = best;
    si[tid] = bi;
    __syncthreads();

    for (int s = 128; s > 0; s >>= 1) {
        if (tid < s) {
            const float v2 = sv[tid + s];
            const int   i2 = si[tid + s];
            if (v2 > sv[tid] || (v2 == sv[tid] && i2 < si[tid])) {
                sv[tid] = v2;
                si[tid] = i2;
            }
        }
        __syncthreads();
    }

    if (tid == 0) {
        const int idx = si[0];
        const int r_index = idx / LEN_R;
        const int d_index = idx - (idx / LEN_R) * LEN_D;   // == idx % 64
        const float rv = truncf(r_t[r_index]);
        const float dv = truncf(d_t[d_index]);
        tail[b]           = dv;   // d
        tail[BATCH + b]   = rv;   // r
        tail[2*BATCH + b] = dv;   // d
    }
}

extern "C" void kernel_launch(void* const* d_in, const int* in_sizes, int n_in,
                              void* d_out, int out_size, void* d_ws, size_t ws_size,
                              hipStream_t stream) {
    const float* x   = (const float*)d_in[0];   // [256,8,16,4096]
    const float* W   = (const float*)d_in[1];   // [1,8,16,1] -> 128 contiguous
    const float* r_t = (const float*)d_in[2];   // [64]
    const float* d_t = (const float*)d_in[3];   // [64]
    float* out = (float*)d_out;

    dim3 grid1(NDIM / (8 * 16), BATCH);         // (32, 256)
    conv_einsum_wmma_kernel<<<grid1, 256, 0, stream>>>(x, W, out);

    float* tail = out + (size_t)BATCH * NDIM;   // after outputs block
    argmax_rd_kernel<<<BATCH, 256, 0, stream>>>(out, r_t, d_t, tail);
}